// TopKWrapper_58609123721987
// MI455X (gfx1250) — compile-verified
//
#include <hip/hip_runtime.h>
#include <stdint.h>

typedef __attribute__((ext_vector_type(2))) float v2f;
typedef __attribute__((ext_vector_type(4))) float v4f;
typedef __attribute__((ext_vector_type(8))) float v8f;
typedef __attribute__((ext_vector_type(4))) int   v4i;

// builtin-required pointer flavors: AS(1) = global, AS(3) = LDS, pointee = int4 (b128)
typedef __attribute__((address_space(1))) v4i* gbl_v4i_p;
typedef __attribute__((address_space(3))) v4i* lds_v4i_p;

#if __has_builtin(__builtin_amdgcn_global_load_async_to_lds_b128) && \
    __has_builtin(__builtin_amdgcn_s_wait_asynccnt)
#define HAVE_ASYNC 1
#else
#define HAVE_ASYNC 0
#endif

#define DIM 256
#define TOPK 10
#define ROWS_PER_BLOCK 64
#define COLS_PER_ITER 32
#define BST 260              // padded LDS stride -> conflict-free fragment loads
#define NEG_INF (-3.4e38f)

// 16-byte global -> LDS copy: async DMA on CDNA5, sync fallback otherwise.
__device__ __forceinline__ void copy16_to_lds(const float* g, float* l) {
#if HAVE_ASYNC
    __builtin_amdgcn_global_load_async_to_lds_b128(
        (gbl_v4i_p)(uintptr_t)g,
        (lds_v4i_p)(uint32_t)(uintptr_t)l,
        0, 0);
#else
    *(v4f*)l = *(const v4f*)g;
#endif
}

__device__ __forceinline__ void wait_async_copies() {
#if HAVE_ASYNC
    __builtin_amdgcn_s_wait_asynccnt(0);
#endif
}

#define WMMA_F32X4(A, B, C) \
    __builtin_amdgcn_wmma_f32_16x16x4_f32(false, (A), false, (B), (short)0, (C), false, false)

__global__ __launch_bounds__(256, 1)
void topk_gemm_wmma_kernel(const float* __restrict__ img,
                           const float* __restrict__ txt,
                           int* __restrict__ out,
                           int N, int M)
{
    __shared__ float B_lds[2][COLS_PER_ITER * BST];        //  66,560 B (double buffer)
    __shared__ float s_listv[8][16][TOPK];
    __shared__ int   s_listi[8][16][TOPK];
    __shared__ float s_candv[8][16][16];
    __shared__ int   s_candi[8][16][16];
    __shared__ int   s_cnt[8][16];
    __shared__ float s_thr[8][16];

    const int tid  = threadIdx.x;
    const int wave = tid >> 5;     // 0..7
    const int lane = tid & 31;
    const int hi   = lane >> 4;
    const int nloc = lane & 15;
    const int rt   = wave >> 1;    // row tile 0..3
    const int ch   = wave & 1;     // column half 0..1

    const int row0 = blockIdx.x * ROWS_PER_BLOCK;

    // ---- init per-wave top-k state ----
    if (tid < 128) {
        int w = tid >> 4, r = tid & 15;
        s_thr[w][r] = NEG_INF;
        s_cnt[w][r] = 0;
        for (int i = 0; i < TOPK; ++i) { s_listv[w][r][i] = NEG_INF; s_listi[w][r][i] = 0; }
    }

    // ---- stage B tile 0 (32 x 256) via async DMA ----
    {
        int kk = (tid & 63) * 4;
        int g  = tid >> 6;
        for (int p = 0; p < 8; ++p) {
            int c = p * 4 + g;
            copy16_to_lds(&txt[(size_t)c * DIM + kk], &B_lds[0][c * BST + kk]);
        }
    }

    // ---- A operand lives in registers for the whole kernel ----
    // WMMA f32 A layout: lane holds row (nloc), K pair at k0 + 2*hi.
    // 64 x v2f = 128 VGPRs/lane; reused across all M tiles -> halves LDS traffic.
    v2f a_frag[DIM / 4];
    {
        const float* arow = &img[(size_t)(row0 + rt * 16 + nloc) * DIM + 2 * hi];
        #pragma unroll
        for (int q = 0; q < DIM / 4; ++q)
            a_frag[q] = *(const v2f*)&arow[q * 4];
    }

    wait_async_copies();
    __syncthreads();

    // B fragment base offset (+k0 added inside the K loop)
    const int b_off = (ch * 16 + nloc) * BST + 2 * hi;

    const int mtiles = M / COLS_PER_ITER;
    for (int mt = 0; mt < mtiles; ++mt) {
        const int colb = mt * COLS_PER_ITER;
        const int cur  = mt & 1;

        // ---- issue async staging of the NEXT B tile into the other buffer ----
        if (mt + 1 < mtiles) {
            int kk = (tid & 63) * 4;
            int g  = tid >> 6;
            for (int p = 0; p < 8; ++p) {
                int c = p * 4 + g;
                copy16_to_lds(&txt[(size_t)(colb + COLS_PER_ITER + c) * DIM + kk],
                              &B_lds[cur ^ 1][c * BST + kk]);
            }
        }
        // L2 prefetch two tiles ahead (global_prefetch_b8)
        if (mt + 2 < mtiles) {
            __builtin_prefetch(&txt[(size_t)(colb + 2 * COLS_PER_ITER + (tid >> 3)) * DIM
                                    + (size_t)(tid & 7) * 32], 0, 1);
        }

        // ---- 16x16 tile GEMM over K=256: 64 WMMAs, 4 independent chains,
        //      A from registers, only B streams from LDS ----
        const float* Bb = B_lds[cur];
        v8f acc0 = {}, acc1 = {}, acc2 = {}, acc3 = {};
        #pragma unroll
        for (int k0 = 0; k0 < DIM; k0 += 16) {
            v2f b0 = *(const v2f*)&Bb[b_off + k0];
            v2f b1 = *(const v2f*)&Bb[b_off + k0 + 4];
            v2f b2 = *(const v2f*)&Bb[b_off + k0 + 8];
            v2f b3 = *(const v2f*)&Bb[b_off + k0 + 12];
            acc0 = WMMA_F32X4(a_frag[k0 / 4 + 0], b0, acc0);
            acc1 = WMMA_F32X4(a_frag[k0 / 4 + 1], b1, acc1);
            acc2 = WMMA_F32X4(a_frag[k0 / 4 + 2], b2, acc2);
            acc3 = WMMA_F32X4(a_frag[k0 / 4 + 3], b3, acc3);
        }
        v8f acc = (acc0 + acc1) + (acc2 + acc3);

        // ---- threshold filter: vectorized threshold read, rare candidate push ----
        v4f t0 = *(const v4f*)&s_thr[wave][hi * 8];
        v4f t1 = *(const v4f*)&s_thr[wave][hi * 8 + 4];
        const int colg = colb + ch * 16 + nloc;
        bool pushed = false;
        #pragma unroll
        for (int j = 0; j < 8; ++j) {
            float th = (j < 4) ? t0[j] : t1[j - 4];
            float v  = acc[j];
            if (v > th) {
                int r = hi * 8 + j;               // C layout: VGPR j holds row j + 8*hi
                int p = atomicAdd(&s_cnt[wave][r], 1);   // <= 16 per row per tile
                s_candv[wave][r][p] = v;
                s_candi[wave][r][p] = colg;
                pushed = true;
            }
        }

        // ---- wave-private candidate processing (skipped when no pushes) ----
        // candidates/lists/thresholds are wave-local; DS ops in a wave are in-order.
        if (__any((int)pushed)) {
            if (lane < 16) {
                int r = lane;
                int n = s_cnt[wave][r];
                for (int i = 0; i < n; ++i) {
                    float v  = s_candv[wave][r][i];
                    int   ix = s_candi[wave][r][i];
                    if (v > s_listv[wave][r][TOPK - 1]) {
                        int p = TOPK - 1;
                        while (p > 0 && v > s_listv[wave][r][p - 1]) {
                            s_listv[wave][r][p] = s_listv[wave][r][p - 1];
                            s_listi[wave][r][p] = s_listi[wave][r][p - 1];
                            --p;
                        }
                        s_listv[wave][r][p] = v;
                        s_listi[wave][r][p] = ix;
                    }
                }
                s_cnt[wave][r] = 0;
                s_thr[wave][r] = s_listv[wave][r][TOPK - 1];
            }
        }

        // next B tile must have landed; one block barrier per iteration
        wait_async_copies();
        __syncthreads();
    }

    // ---- merge the two column-half lists per row, write indices ----
    if (tid < ROWS_PER_BLOCK) {
        int rl = tid & 15;
        int wa = (tid >> 4) * 2, wb = wa + 1;
        int pa = 0, pb = 0;
        int* o = &out[(size_t)(row0 + tid) * TOPK];
        for (int i = 0; i < TOPK; ++i) {
            float va = s_listv[wa][rl][pa], vb = s_listv[wb][rl][pb];
            int   ia = s_listi[wa][rl][pa], ib = s_listi[wb][rl][pb];
            bool  ta = (va > vb) || (va == vb && ia <= ib);
            o[i] = ta ? ia : ib;
            if (ta) ++pa; else ++pb;
        }
    }
}

extern "C" void kernel_launch(void* const* d_in, const int* in_sizes, int n_in,
                              void* d_out, int out_size, void* d_ws, size_t ws_size,
                              hipStream_t stream) {
    const float* img = (const float*)d_in[0];
    const float* txt = (const float*)d_in[1];
    int* out = (int*)d_out;
    int N = in_sizes[0] / DIM;   // 8192
    int M = in_sizes[1] / DIM;   // 32768
    dim3 grid(N / ROWS_PER_BLOCK), block(256);
    topk_gemm_wmma_kernel<<<grid, block, 0, stream>>>(img, txt, out, N, M);
}